// GAE_25890062861058
// MI455X (gfx1250) — compile-verified
//
#include <hip/hip_runtime.h>
#include <math.h>

typedef float v2f __attribute__((ext_vector_type(2)));
typedef float v8f __attribute__((ext_vector_type(8)));

#define N_NODES 10000
#define N_EDGES 320000
#define CIN     128
#define HDIM    256
#define KPOOL   5000

// ---------------- utility kernels ----------------

__global__ void zero_f(float* __restrict__ p, int n) {
    int i = blockIdx.x * blockDim.x + threadIdx.x;
    if (i < n) p[i] = 0.f;
}

__global__ void i2f_copy(const int* __restrict__ in, float* __restrict__ out, int n) {
    int i = blockIdx.x * blockDim.x + threadIdx.x;
    if (i < n) out[i] = (float)in[i];
}

// ---------------- WMMA fp32 GEMM ----------------
// C[M,Nd] = op(A[M,Kd] @ B[Kd,Nd] + bias + res), row-major, per-wave 16x64 tile.
// Uses V_WMMA_F32_16X16X4_F32: A frag = 16x4 (2 VGPRs), B frag = 4x16 (2 VGPRs),
// C/D = 16x16 f32 in 8 VGPRs.  Kd % 4 == 0, M % 16 == 0, Nd % 64 == 0.
__global__ void wmma_gemm_f32(const float* __restrict__ A, int lda,
                              const float* __restrict__ B, int ldb,
                              const float* __restrict__ bias,
                              const float* __restrict__ res, int ldres,
                              float* __restrict__ C, int ldc,
                              int M, int Kd, int Nd, int relu)
{
    const int wavesPerBlk = blockDim.x >> 5;
    const int w = blockIdx.x * wavesPerBlk + (threadIdx.x >> 5);
    const int mTiles  = M >> 4;
    const int nGroups = Nd >> 6;
    if (w >= mTiles * nGroups) return;          // wave-uniform exit: EXEC stays all-1s
    const int tm = (w % mTiles) << 4;
    const int tn = (w / mTiles) << 6;

    const int lane = threadIdx.x & 31;
    const int half = lane >> 4;                 // 0: K pair {0,1}, 1: K pair {2,3}
    const int lid  = lane & 15;                 // M row (A) / N col (B,C)

    v8f acc[4] = {};

    const float* Arow = A + (long)(tm + lid) * lda;
    for (int k = 0; k < Kd; k += 4) {
        v2f a;
        a.x = Arow[k + 2 * half + 0];
        a.y = Arow[k + 2 * half + 1];
        const float* Bk0 = B + (long)(k + 2 * half + 0) * ldb + tn;
        const float* Bk1 = B + (long)(k + 2 * half + 1) * ldb + tn;
#pragma unroll
        for (int t = 0; t < 4; ++t) {
            v2f b;
            b.x = Bk0[t * 16 + lid];
            b.y = Bk1[t * 16 + lid];
            acc[t] = __builtin_amdgcn_wmma_f32_16x16x4_f32(
                false, a, false, b, (short)0, acc[t], false, false);
        }
    }

#pragma unroll
    for (int t = 0; t < 4; ++t) {
        const int col = tn + t * 16 + lid;
#pragma unroll
        for (int r = 0; r < 8; ++r) {
            const int row = tm + half * 8 + r;
            float v = acc[t][r];
            if (bias) v += bias[col];
            if (res)  v += res[(long)row * ldres + col];
            if (relu) v = v > 0.f ? v : 0.f;
            C[(long)row * ldc + col] = v;
        }
    }
}

// ---------------- graph scatter kernels ----------------

// agg[dst[e], :] += y[src[e], :]   (4 floats per thread)
__global__ void scatter_add_rows(const float* __restrict__ y,
                                 const int* __restrict__ src,
                                 const int* __restrict__ dst,
                                 float* __restrict__ agg, int E, int Hh)
{
    const int chunks = Hh >> 2;
    int i = blockIdx.x * blockDim.x + threadIdx.x;
    if (i >= E * chunks) return;
    int e = i / chunks;
    int c = (i % chunks) << 2;
    const float* s = y + (long)src[e] * Hh + c;
    float* d = agg + (long)dst[e] * Hh + c;
    atomicAdd(d + 0, s[0]);
    atomicAdd(d + 1, s[1]);
    atomicAdd(d + 2, s[2]);
    atomicAdd(d + 3, s[3]);
}

// per-node dual dot products: score[i] = x1[i].w_self ; tbuf[i] = x1[i].w_msg
__global__ void score_nodes(const float* __restrict__ x1,
                            const float* __restrict__ wself,
                            const float* __restrict__ wmsg,
                            float* __restrict__ score, float* __restrict__ tbuf, int n)
{
    int node = blockIdx.x * (blockDim.x >> 5) + (threadIdx.x >> 5);
    if (node >= n) return;
    int lane = threadIdx.x & 31;
    float s = 0.f, m = 0.f;
    const float* row = x1 + (long)node * (2 * HDIM);
    for (int c = lane; c < 2 * HDIM; c += 32) {
        float v = row[c];
        s += v * wself[c];
        m += v * wmsg[c];
    }
    for (int off = 16; off > 0; off >>= 1) {
        s += __shfl_xor(s, off);
        m += __shfl_xor(m, off);
    }
    if (lane == 0) { score[node] = s; tbuf[node] = m; }
}

__global__ void score_edges(const float* __restrict__ tbuf,
                            const int* __restrict__ src,
                            const int* __restrict__ dst,
                            float* __restrict__ score, int E)
{
    int e = blockIdx.x * blockDim.x + threadIdx.x;
    if (e < E) atomicAdd(&score[dst[e]], tbuf[src[e]]);
}

// descending rank with index tie-break (matches jax.lax.top_k ordering)
__global__ void rank_topk(const float* __restrict__ score,
                          float* __restrict__ ssort, int* __restrict__ perm,
                          int n, int k)
{
    int i = blockIdx.x * blockDim.x + threadIdx.x;
    if (i >= n) return;
    float si = score[i];
    int r = 0;
    for (int j = 0; j < n; ++j) {
        float sj = score[j];
        r += (sj > si) || (sj == si && j < i);
    }
    if (r < k) { perm[r] = i; ssort[r] = si; }
}

__global__ void init_nmap(int* __restrict__ nmap, int n, int k) {
    int i = blockIdx.x * blockDim.x + threadIdx.x;
    if (i < n) nmap[i] = k;
}

__global__ void set_nmap(const int* __restrict__ perm, int* __restrict__ nmap, int k) {
    int i = blockIdx.x * blockDim.x + threadIdx.x;
    if (i < k) nmap[perm[i]] = i;
}

__global__ void apool_kernel(const int* __restrict__ edge, const int* __restrict__ nmap,
                             float* __restrict__ out, int n) {
    int i = blockIdx.x * blockDim.x + threadIdx.x;
    if (i < n) out[i] = (float)nmap[edge[i]];
}

// x_pool[k,c] = x1[perm[k],c] * tanh(ssort[k]); also scatter into xcat unpool slot
__global__ void pool_scatter(const float* __restrict__ x1,
                             const int* __restrict__ perm,
                             const float* __restrict__ ssort,
                             float* __restrict__ xpool_out,
                             float* __restrict__ xcat, int k)
{
    int i = blockIdx.x * blockDim.x + threadIdx.x;
    if (i >= k * 2 * HDIM) return;
    int kk = i / (2 * HDIM);
    int c  = i % (2 * HDIM);
    float g = tanhf(ssort[kk]);
    float v = x1[(long)perm[kk] * (2 * HDIM) + c] * g;
    xpool_out[i] = v;
    xcat[(long)perm[kk] * (3 * HDIM) + HDIM + c] = v;
}

// ---------------- host side ----------------

static inline int cdiv(long a, long b) { return (int)((a + b - 1) / b); }

extern "C" void kernel_launch(void* const* d_in, const int* in_sizes, int n_in,
                              void* d_out, int out_size, void* d_ws, size_t ws_size,
                              hipStream_t stream) {
    const float* x        = (const float*)d_in[0];
    const int*   ei       = (const int*)  d_in[1];
    const float* W_pre    = (const float*)d_in[4];
    const float* b_pre    = (const float*)d_in[5];
    const float* W1_self  = (const float*)d_in[6];
    const float* W1_msg   = (const float*)d_in[7];
    const float* b1       = (const float*)d_in[8];
    const float* w_sself  = (const float*)d_in[9];
    const float* w_smsg   = (const float*)d_in[10];
    const float* W2_self  = (const float*)d_in[11];
    const float* W2_msg   = (const float*)d_in[12];
    const float* b2       = (const float*)d_in[13];
    const float* W_post   = (const float*)d_in[14];
    const float* b_post   = (const float*)d_in[15];

    const int* src = ei;
    const int* dst = ei + N_EDGES;

    // workspace layout (floats)
    float* ws    = (float*)d_ws;
    float* x1    = ws;                         // [N, 2H]   h1 | x0
    float* xcat  = ws + 5120000;               // [N, 3H]   h2 | x_unpool
    float* ybuf  = ws + 12800000;              // [N, H]
    float* agg   = ws + 15360000;              // [N, H]
    float* score = ws + 17920000;              // [N]
    float* tbuf  = ws + 17930000;              // [N]
    float* ssort = ws + 17940000;              // [K]
    int*   perm  = (int*)(ws + 17945000);      // [K]
    int*   nmap  = (int*)(ws + 17950000);      // [N]

    // output layout (flat, return order)
    float* out_main  = (float*)d_out;                 // [N, CIN]
    float* out_edge  = out_main + (long)N_NODES*CIN;  // [2, E]
    float* out_xpool = out_edge + 2L*N_EDGES;         // [K, 2H]
    float* out_apool = out_xpool + (long)KPOOL*2*HDIM;// [2, E]
    float* out_perm  = out_apool + 2L*N_EDGES;        // [K]

    const int B = 256;
    auto gemm = [&](const float* A, int lda, const float* Bm, int ldb,
                    const float* bias, const float* res, int ldres,
                    float* C, int ldc, int M, int Kd, int Nd, int relu) {
        int waves  = (M / 16) * (Nd / 64);
        int blocks = cdiv((long)waves * 32, B);
        wmma_gemm_f32<<<blocks, B, 0, stream>>>(A, lda, Bm, ldb, bias, res, ldres,
                                                C, ldc, M, Kd, Nd, relu);
    };

    // 1) pre_mlp: x0 = relu(x @ W_pre + b_pre)  -> x1[:, H:2H]
    gemm(x, CIN, W_pre, HDIM, b_pre, nullptr, 0,
         x1 + HDIM, 2 * HDIM, N_NODES, CIN, HDIM, 1);

    // 2) conv1 message transform: y = x0 @ W1_msg
    gemm(x1 + HDIM, 2 * HDIM, W1_msg, HDIM, nullptr, nullptr, 0,
         ybuf, HDIM, N_NODES, HDIM, HDIM, 0);

    // 3) agg1 = scatter_add(y[src] -> dst)
    zero_f<<<cdiv((long)N_NODES * HDIM, B), B, 0, stream>>>(agg, N_NODES * HDIM);
    scatter_add_rows<<<cdiv((long)N_EDGES * (HDIM / 4), B), B, 0, stream>>>(
        ybuf, src, dst, agg, N_EDGES, HDIM);

    // 4) h1 = relu(x0 @ W1_self + agg + b1) -> x1[:, 0:H]
    gemm(x1 + HDIM, 2 * HDIM, W1_self, HDIM, b1, agg, HDIM,
         x1, 2 * HDIM, N_NODES, HDIM, HDIM, 1);

    // 5) scores
    score_nodes<<<cdiv(N_NODES, B / 32), B, 0, stream>>>(x1, w_sself, w_smsg,
                                                         score, tbuf, N_NODES);
    score_edges<<<cdiv(N_EDGES, B), B, 0, stream>>>(tbuf, src, dst, score, N_EDGES);

    // 6) top-k via O(N^2) stable descending rank
    rank_topk<<<cdiv(N_NODES, B), B, 0, stream>>>(score, ssort, perm, N_NODES, KPOOL);

    // 7) pooling + unpool scatter (xcat[:, H:3H] zero-initialized first)
    zero_f<<<cdiv((long)N_NODES * 3 * HDIM, B), B, 0, stream>>>(xcat, N_NODES * 3 * HDIM);
    pool_scatter<<<cdiv((long)KPOOL * 2 * HDIM, B), B, 0, stream>>>(
        x1, perm, ssort, out_xpool, xcat, KPOOL);

    // 8) node map + integer-ish outputs (as float)
    init_nmap<<<cdiv(N_NODES, B), B, 0, stream>>>(nmap, N_NODES, KPOOL);
    set_nmap<<<cdiv(KPOOL, B), B, 0, stream>>>(perm, nmap, KPOOL);
    apool_kernel<<<cdiv(2L * N_EDGES, B), B, 0, stream>>>(ei, nmap, out_apool, 2 * N_EDGES);
    i2f_copy<<<cdiv(2L * N_EDGES, B), B, 0, stream>>>(ei, out_edge, 2 * N_EDGES);
    i2f_copy<<<cdiv(KPOOL, B), B, 0, stream>>>(perm, out_perm, KPOOL);

    // 9) conv2 message transform: y = x_unpool @ W2_msg   (x_unpool = xcat[:, H:3H])
    gemm(xcat + HDIM, 3 * HDIM, W2_msg, HDIM, nullptr, nullptr, 0,
         ybuf, HDIM, N_NODES, 2 * HDIM, HDIM, 0);

    // 10) agg2
    zero_f<<<cdiv((long)N_NODES * HDIM, B), B, 0, stream>>>(agg, N_NODES * HDIM);
    scatter_add_rows<<<cdiv((long)N_EDGES * (HDIM / 4), B), B, 0, stream>>>(
        ybuf, src, dst, agg, N_EDGES, HDIM);

    // 11) h2 = relu(x_unpool @ W2_self + agg + b2) -> xcat[:, 0:H]
    gemm(xcat + HDIM, 3 * HDIM, W2_self, HDIM, b2, agg, HDIM,
         xcat, 3 * HDIM, N_NODES, 2 * HDIM, HDIM, 1);

    // 12) post_mlp: out = xcat @ W_post + b_post
    gemm(xcat, 3 * HDIM, W_post, CIN, b_post, nullptr, 0,
         out_main, CIN, N_NODES, 3 * HDIM, CIN, 0);
}